// GlobalPointer_63582695850575
// MI455X (gfx1250) — compile-verified
//
#include <hip/hip_runtime.h>
#include <hip/hip_bf16.h>
#include <math.h>

// Problem constants
#define BATCH   8
#define SEQLEN  1024
#define HID     1024
#define HEADS   12
#define HEADDIM 64
#define NCOLS   (HEADS * HEADDIM * 2)   // 1536
#define NEG_BIG 1.0e12f

typedef __attribute__((ext_vector_type(8))) float v8f;
typedef __attribute__((ext_vector_type(2))) float v2f;

// ---------------------------------------------------------------------------
// Kernel 1: seq = X @ W + b  ->  RoPE  ->  Q,(B,H,L,D) and K,(B,H,L,D) in ws.
// Grid: x = NCOLS/16 = 96 (j-tiles), y = M/64 = 128 (m-blocks). Block = 128.
// Each wave computes one 16x16 (m,j) tile with V_WMMA_F32_16X16X4_F32,
// W tile staged in LDS and shared across the 4 waves.
// ---------------------------------------------------------------------------
__global__ __launch_bounds__(128) void proj_rope_kernel(
    const float* __restrict__ X, const float* __restrict__ W,
    const float* __restrict__ bias, float* __restrict__ Q,
    float* __restrict__ Kb) {
  __shared__ float wlds[64][20];  // 64 k-rows x 16 j-cols, stride 20 (bank pad)

  const int t    = threadIdx.x;
  const int wave = t >> 5;
  const int lane = t & 31;
  const int cl   = lane & 15;       // column within tile / M row within half
  const int hi   = lane >> 4;       // 0: K0,K1 ; 1: K2,K3 (A & B layouts)
  const int j0   = blockIdx.x * 16;
  const int m0   = blockIdx.y * 64 + wave * 16;

  // A operand: per-lane row pointer into X (16x4 f32 A layout)
  const float* xrow = X + (size_t)(m0 + cl) * HID + hi * 2;

  // cooperative W staging: thread t loads 8 floats of k-row (t>>1)
  const int srow = t >> 1;
  const int scs  = (t & 1) * 8;
  const float* wsrc = W + (size_t)srow * NCOLS + j0 + scs;

  v8f c = {};
  for (int k0 = 0; k0 < HID; k0 += 64) {
    float4 w0 = *(const float4*)(wsrc);
    float4 w1 = *(const float4*)(wsrc + 4);
    wsrc += (size_t)64 * NCOLS;
    __syncthreads();  // previous chunk fully consumed
    *(float4*)&wlds[srow][scs]     = w0;
    *(float4*)&wlds[srow][scs + 4] = w1;
    __syncthreads();  // chunk staged
#pragma unroll
    for (int kk = 0; kk < 64; kk += 4) {
      v2f a = *(const v2f*)(xrow + k0 + kk);
      v2f b;
      b.x = wlds[kk + hi * 2][cl];
      b.y = wlds[kk + hi * 2 + 1][cl];
      c = __builtin_amdgcn_wmma_f32_16x16x4_f32(
          /*neg_a=*/false, a, /*neg_b=*/false, b,
          /*c_mod=*/(short)0, c, /*reuse_a=*/false, /*reuse_b=*/false);
    }
  }

  // Epilogue: bias + RoPE + scatter to Q/K (both (B,H,L,D), coalesced).
  const int   j    = j0 + cl;
  const float bj   = bias[j];
  const int   h    = j >> 7;            // head
  const int   cc   = j & 127;           // channel within head pair-block
  const bool  isq  = cc < HEADDIM;
  const int   d    = isq ? cc : (cc - HEADDIM);
  const int   i2   = d & ~1;            // 2i
  // inv_freq = 10000^(-(2i)/64) = exp(-(2i) * ln(10000)/64)
  const float invf = __expf(-(float)i2 * 0.14391156516f);
  const float sgn  = (d & 1) ? 1.0f : -1.0f;  // even: v*c - vp*s ; odd: v*c + vp*s
  const int   bidx = m0 >> 10;          // batch (tile never straddles batches)
  float* dst = (isq ? Q : Kb) + ((size_t)(bidx * HEADS + h) * SEQLEN) * HEADDIM + d;

#pragma unroll
  for (int r = 0; r < 8; ++r) {
    const int m = m0 + r + (hi << 3);   // C layout: VGPR r -> rows r / r+8
    const int l = m & (SEQLEN - 1);
    float sn, cs;
    sincosf((float)l * invf, &sn, &cs);
    const float v  = c[r] + bj;
    const float vp = __shfl_xor(v, 1, 32);  // pair partner (col ^ 1)
    dst[(size_t)l * HEADDIM] = v * cs + sgn * sn * vp;
  }
}

// ---------------------------------------------------------------------------
// Kernel 2: logits[b,h,m,n] = masked/causal/scaled Q K^T.
// Grid: x = L/16 = 64 (n-tiles), y = L/64 = 16 (m-blocks), z = B*H = 96.
// Block = 128 (4 waves); K tile (16x64) staged in LDS, shared by waves.
// Strictly-lower tiles emit the exact f32 constant (s - 1e12 == -1e12 in f32).
// ---------------------------------------------------------------------------
__global__ __launch_bounds__(128) void qk_logits_kernel(
    const float* __restrict__ Q, const float* __restrict__ Kb,
    const int* __restrict__ mask, float* __restrict__ out) {
  __shared__ float klds[16][68];  // 16 n-rows x 64 k, stride 68 (bank pad)

  const int t    = threadIdx.x;
  const int wave = t >> 5;
  const int lane = t & 31;
  const int cl   = lane & 15;
  const int hi   = lane >> 4;
  const int bh   = blockIdx.z;
  const int b    = bh / HEADS;
  const int n0   = blockIdx.x * 16;
  const int m0b  = blockIdx.y * 64;
  const int m0   = m0b + wave * 16;

  const int  n    = n0 + cl;
  const int  padn = mask[b * SEQLEN + n];   // 0 => padded
  const size_t obase = (size_t)bh * SEQLEN * SEQLEN;

  const bool blockLower = (m0b > n0 + 15);  // whole block strictly below diag
  const bool fullLower  = (m0  > n0 + 15);  // this wave's tile strictly below

  if (!blockLower) {
    // stage K[n0..n0+15][0..63] into LDS
    const int   krow = t >> 3;
    const int   kcs  = (t & 7) * 8;
    const float* ks  = Kb + ((size_t)bh * SEQLEN + n0 + krow) * HEADDIM + kcs;
    float4 k0v = *(const float4*)(ks);
    float4 k1v = *(const float4*)(ks + 4);
    *(float4*)&klds[krow][kcs]     = k0v;
    *(float4*)&klds[krow][kcs + 4] = k1v;
    __syncthreads();
  }

  v8f c = {};
  if (!fullLower) {
    const float* qrow = Q + ((size_t)bh * SEQLEN + m0 + cl) * HEADDIM + hi * 2;
#pragma unroll
    for (int kk = 0; kk < HEADDIM; kk += 4) {
      v2f a = *(const v2f*)(qrow + kk);
      v2f bfr;
      bfr.x = klds[cl][kk + hi * 2];      // B[k][n] = K[n][k]
      bfr.y = klds[cl][kk + hi * 2 + 1];
      c = __builtin_amdgcn_wmma_f32_16x16x4_f32(
          false, a, false, bfr, (short)0, c, false, false);
    }
  }

#pragma unroll
  for (int r = 0; r < 8; ++r) {
    const int m    = m0 + r + (hi << 3);
    const int padm = mask[b * SEQLEN + m];
    float v;
    if (fullLower) {
      v = -1.25e11f;                      // (s - 1e12)/8 rounds to this exactly
    } else {
      v = c[r];
      if (m > n) v -= NEG_BIG;
      v *= 0.125f;                        // 1/sqrt(64)
    }
    if (!padm || !padn) v = -__builtin_inff();
    __builtin_nontemporal_store(v, &out[obase + (size_t)m * SEQLEN + n]);
  }
}

// ---------------------------------------------------------------------------
extern "C" void kernel_launch(void* const* d_in, const int* in_sizes, int n_in,
                              void* d_out, int out_size, void* d_ws,
                              size_t ws_size, hipStream_t stream) {
  const float* X    = (const float*)d_in[0];  // (8,1024,1024) f32
  const int*   mask = (const int*)d_in[1];    // (8,1024) i32
  const float* W    = (const float*)d_in[2];  // (1024,1536) f32
  const float* bias = (const float*)d_in[3];  // (1536,) f32
  float*       out  = (float*)d_out;          // (8,12,1024,1024) f32

  float* qbuf = (float*)d_ws;                                   // 25.2 MB
  float* kbuf = qbuf + (size_t)BATCH * HEADS * SEQLEN * HEADDIM; // 25.2 MB

  dim3 blk(128);
  dim3 g1(NCOLS / 16, (BATCH * SEQLEN) / 64, 1);   // 96 x 128
  proj_rope_kernel<<<g1, blk, 0, stream>>>(X, W, bias, qbuf, kbuf);

  dim3 g2(SEQLEN / 16, SEQLEN / 64, BATCH * HEADS); // 64 x 16 x 96
  qk_logits_kernel<<<g2, blk, 0, stream>>>(qbuf, kbuf, mask, out);
}